// SpectralBatchNorm_34359738791
// MI455X (gfx1250) — compile-verified
//
#include <hip/hip_runtime.h>
#include <hip/hip_bf16.h>

typedef float v2f __attribute__((ext_vector_type(2)));
typedef float v8f __attribute__((ext_vector_type(8)));

#define BATCH 256
#define SEQ   1024
#define HID   256
#define EPS   1e-5f
#define NS_ITERS 6
#define MAT   (HID*HID)                  // 65536 floats per per-s matrix
#define XCH   ((size_t)SEQ*(size_t)MAT)  // per full batched matrix

// ---------------- mean over batch: mean[s][h] = (1/B) sum_b x[b][s][h] -------
__global__ void mean_kernel(const float* __restrict__ x, float* __restrict__ mean) {
    const int s = blockIdx.x, h = threadIdx.x;
    const float* p = x + (size_t)s * HID + h;
    float sum = 0.f;
#pragma unroll 4
    for (int b = 0; b < BATCH; ++b) sum += p[(size_t)b * SEQ * HID];
    mean[s * HID + h] = sum * (1.0f / BATCH);
}

// ---------------- center: xc[s][b][h] = x[b][s][h] - mean[s][h] --------------
__global__ void center_kernel(const float* __restrict__ x, const float* __restrict__ mean,
                              float* __restrict__ xc) {
    const int s = blockIdx.x, b = blockIdx.y, h = threadIdx.x;
    xc[(size_t)s * MAT + (size_t)b * HID + h] =
        x[(size_t)b * SEQ * HID + (size_t)s * HID + h] - mean[s * HID + h];
}

// ---------------- batched 256x256x256 GEMM via v_wmma_f32_16x16x4_f32 --------
// C[s] = alpha*(scaleS?scaleS[s]:1) * op(A[s]) * B[s]  (+ beta * D[s] if D)
// op(A) = A or A^T (transA).  A,B,D are 256x256 row-major, stride MAT per s.
// C has arbitrary row stride ldC and batch stride strideC (lets us write y
// directly into the (b,s,h)-layout output).
__global__ __launch_bounds__(256) void gemm256_wmma(
    const float* __restrict__ Am, int transA,
    const float* __restrict__ Bm,
    float* __restrict__ Cm, long long ldC, long long strideC,
    const float* __restrict__ Dm, float alpha, float beta,
    const float* __restrict__ scaleS)
{
    __shared__ float ldsA[128 * 68];   // !transA: [m][k] rows of 68 (pad 4)
                                       //  transA: [k][m] rows of 132 (pad 4)
    __shared__ float ldsB[64 * 132];   // [k][n] rows of 132 (pad 4)

    const int s = blockIdx.z;
    const int rowBase = blockIdx.x * 128;
    const int colBase = blockIdx.y * 128;
    const int tid  = threadIdx.x;
    const int lane = tid & 31, l15 = lane & 15, hi = lane >> 4;
    const int w  = tid >> 5;           // wave id 0..7 (wave32)
    const int wr = w & 3, wc = w >> 2; // 4x2 wave grid, each wave 32x64 of C

    const float* Ab = Am + (size_t)s * MAT;
    const float* Bb = Bm + (size_t)s * MAT;

    v8f acc[2][4];
#pragma unroll
    for (int i = 0; i < 2; ++i)
#pragma unroll
        for (int j = 0; j < 4; ++j)
#pragma unroll
            for (int r = 0; r < 8; ++r) acc[i][j][r] = 0.f;

    for (int k0 = 0; k0 < HID; k0 += 64) {
        // stage B panel: 64 rows(k) x 128 cols(n), float4 cooperative loads
#pragma unroll
        for (int p = 0; p < 8; ++p) {
            int idx = p * 256 + tid;
            int kr = idx >> 5, c4 = idx & 31;
            const float4 v = *(const float4*)(Bb + (size_t)(k0 + kr) * HID + colBase + c4 * 4);
            float* d = &ldsB[kr * 132 + c4 * 4];
            d[0] = v.x; d[1] = v.y; d[2] = v.z; d[3] = v.w;
        }
        if (transA) {
            // stage A^T panel as [k][m]: 64 x 128
#pragma unroll
            for (int p = 0; p < 8; ++p) {
                int idx = p * 256 + tid;
                int kr = idx >> 5, c4 = idx & 31;
                const float4 v = *(const float4*)(Ab + (size_t)(k0 + kr) * HID + rowBase + c4 * 4);
                float* d = &ldsA[kr * 132 + c4 * 4];
                d[0] = v.x; d[1] = v.y; d[2] = v.z; d[3] = v.w;
            }
        } else {
            // stage A panel as [m][k]: 128 x 64
#pragma unroll
            for (int p = 0; p < 8; ++p) {
                int idx = p * 256 + tid;
                int r = idx >> 4, c4 = idx & 15;
                const float4 v = *(const float4*)(Ab + (size_t)(rowBase + r) * HID + k0 + c4 * 4);
                float* d = &ldsA[r * 68 + c4 * 4];
                d[0] = v.x; d[1] = v.y; d[2] = v.z; d[3] = v.w;
            }
        }
        __syncthreads();

        // A-frag (16x4 f32): lanes0-15 hold M=0..15; vgpr0 K=k0+2*hi, vgpr1 +1
        // B-frag (4x16 f32): lanes0-15 hold N=0..15; vgpr0 K=k0+2*hi, vgpr1 +1
        for (int kk = 0; kk < 64; kk += 4) {
            v2f af[2], bf[4];
#pragma unroll
            for (int tr = 0; tr < 2; ++tr) {
                if (transA) {
                    int base = (kk + 2 * hi) * 132 + wr * 32 + tr * 16 + l15;
                    af[tr].x = ldsA[base];
                    af[tr].y = ldsA[base + 132];
                } else {
                    int base = (wr * 32 + tr * 16 + l15) * 68 + kk + 2 * hi;
                    af[tr].x = ldsA[base];
                    af[tr].y = ldsA[base + 1];
                }
            }
#pragma unroll
            for (int tc = 0; tc < 4; ++tc) {
                int base = (kk + 2 * hi) * 132 + wc * 64 + tc * 16 + l15;
                bf[tc].x = ldsB[base];
                bf[tc].y = ldsB[base + 132];
            }
#pragma unroll
            for (int tr = 0; tr < 2; ++tr)
#pragma unroll
                for (int tc = 0; tc < 4; ++tc)
                    acc[tr][tc] = __builtin_amdgcn_wmma_f32_16x16x4_f32(
                        false, af[tr], false, bf[tc], (short)0, acc[tr][tc], false, false);
        }
        __syncthreads();
    }

    const float aEff = alpha * (scaleS ? scaleS[s] : 1.0f);
    float* Cb = Cm + (long long)s * strideC;
    const float* Db = Dm ? (Dm + (size_t)s * MAT) : nullptr;
#pragma unroll
    for (int tr = 0; tr < 2; ++tr)
#pragma unroll
        for (int tc = 0; tc < 4; ++tc) {
            const int m0 = rowBase + wr * 32 + tr * 16;
            const int n0 = colBase + wc * 64 + tc * 16;
#pragma unroll
            for (int r = 0; r < 8; ++r) {
                // C/D f32 16x16 layout: vgpr r => M=r (lanes0-15), M=r+8 (lanes16-31)
                const int m = m0 + r + 8 * hi;
                const int n = n0 + l15;
                float val = aEff * acc[tr][tc][r];
                if (Db) val += beta * Db[(size_t)m * HID + n];
                Cb[(long long)m * ldC + n] = val;
            }
        }
}

// ---------------- trace of (cov + eps*I) per s -------------------------------
__global__ void trace_kernel(const float* __restrict__ A, float* __restrict__ trS,
                             float* __restrict__ invTr, float* __restrict__ rsqTr) {
    __shared__ float red[256];
    const int s = blockIdx.x, t = threadIdx.x;
    red[t] = A[(size_t)s * MAT + (size_t)t * (HID + 1)];
    __syncthreads();
    for (int o = 128; o > 0; o >>= 1) { if (t < o) red[t] += red[t + o]; __syncthreads(); }
    if (t == 0) {
        float tr = red[0] + HID * EPS;
        trS[s] = tr; invTr[s] = 1.0f / tr; rsqTr[s] = rsqrtf(tr);
    }
}

// ------- A := (A + eps*I)/tr  (trace-normalized);  P := I --------------------
__global__ void norm_init_kernel(float* __restrict__ A, float* __restrict__ P,
                                 const float* __restrict__ invTr) {
    const int s = blockIdx.x, i = blockIdx.y, j = threadIdx.x;
    const size_t idx = (size_t)s * MAT + (size_t)i * HID + j;
    float a = A[idx] + (i == j ? EPS : 0.f);
    A[idx] = a * invTr[s];
    P[idx] = (i == j) ? 1.f : 0.f;
}

// ------- lambda_min(cov+epsI) per s via power iteration on B = I - A_n -------
__global__ void mineig_kernel(const float* __restrict__ A, const float* __restrict__ trS,
                              float* __restrict__ minEig) {
    __shared__ float v[256];
    __shared__ float red[256];
    const int s = blockIdx.x, t = threadIdx.x;
    const float* As = A + (size_t)s * MAT + (size_t)t * HID;
    v[t] = 1.0f;
    __syncthreads();
    for (int it = 0; it < 8; ++it) {
        float sum = 0.f;
        for (int j = 0; j < HID; ++j) sum += As[j] * v[j];
        float wi = v[t] - sum;                 // (I - A_n) v
        red[t] = wi * wi; __syncthreads();
        for (int o = 128; o > 0; o >>= 1) { if (t < o) red[t] += red[t + o]; __syncthreads(); }
        float inv = rsqrtf(red[0] + 1e-30f);
        v[t] = wi * inv;
        __syncthreads();
    }
    float sum = 0.f;
    for (int j = 0; j < HID; ++j) sum += As[j] * v[j];
    red[t] = v[t] * (v[t] - sum);              // Rayleigh quotient v^T B v
    __syncthreads();
    for (int o = 128; o > 0; o >>= 1) { if (t < o) red[t] += red[t + o]; __syncthreads(); }
    if (t == 0) minEig[s] = (1.0f - red[0]) * trS[s];
}

// ---------------- global min over s ------------------------------------------
__global__ void minred_kernel(const float* __restrict__ minEig, float* __restrict__ out) {
    __shared__ float red[256];
    const int t = threadIdx.x;
    float m = minEig[t];
    for (int k = 1; k < 4; ++k) m = fminf(m, minEig[t + 256 * k]);
    red[t] = m; __syncthreads();
    for (int o = 128; o > 0; o >>= 1) { if (t < o) red[t] = fminf(red[t], red[t + o]); __syncthreads(); }
    if (t == 0) out[0] = red[0];
}

extern "C" void kernel_launch(void* const* d_in, const int* in_sizes, int n_in,
                              void* d_out, int out_size, void* d_ws, size_t ws_size,
                              hipStream_t stream) {
    const float* x = (const float*)d_in[0];
    float* out = (float*)d_out;
    float* ws = (float*)d_ws;

    size_t off = 0;
    float* mean  = ws + off; off += (size_t)SEQ * HID;
    float* xc    = ws + off; off += XCH;   // centered x, (s,b,h)-contiguous
    float* A     = ws + off; off += XCH;   // cov, then trace-normalized A_n
    float* P     = ws + off; off += XCH;   // Newton-Schulz iterate (ping)
    float* Q     = ws + off; off += XCH;   // Newton-Schulz iterate (pong)
    float* T2    = ws + off; off += XCH;   // temp P*A
    float* T3    = ws + off; off += XCH;   // temp P*(P*A)
    float* trS   = ws + off; off += SEQ;
    float* invTr = ws + off; off += SEQ;
    float* rsqTr = ws + off; off += SEQ;
    float* minE  = ws + off; off += SEQ;

    const dim3 gg(2, 2, SEQ);   // 128x128 C supertiles, one s per blockIdx.z

    mean_kernel<<<SEQ, HID, 0, stream>>>(x, mean);
    center_kernel<<<dim3(SEQ, BATCH), HID, 0, stream>>>(x, mean, xc);

    // cov = (1/B) * xc^T * xc
    gemm256_wmma<<<gg, 256, 0, stream>>>(xc, 1, xc, A, (long long)HID, (long long)MAT,
                                         nullptr, 1.0f / BATCH, 0.f, nullptr);
    trace_kernel<<<SEQ, 256, 0, stream>>>(A, trS, invTr, rsqTr);
    norm_init_kernel<<<dim3(SEQ, HID), HID, 0, stream>>>(A, P, invTr);
    mineig_kernel<<<SEQ, 256, 0, stream>>>(A, trS, minE);

    // Newton-Schulz: P <- 1.5*P - 0.5*P*(P*(P*A_n))
    float* Pc = P;
    float* Qc = Q;
    for (int it = 0; it < NS_ITERS; ++it) {
        gemm256_wmma<<<gg, 256, 0, stream>>>(Pc, 0, A,  T2, (long long)HID, (long long)MAT,
                                             nullptr, 1.0f, 0.f, nullptr);
        gemm256_wmma<<<gg, 256, 0, stream>>>(Pc, 0, T2, T3, (long long)HID, (long long)MAT,
                                             nullptr, 1.0f, 0.f, nullptr);
        gemm256_wmma<<<gg, 256, 0, stream>>>(Pc, 0, T3, Qc, (long long)HID, (long long)MAT,
                                             Pc, -0.5f, 1.5f, nullptr);
        float* t = Pc; Pc = Qc; Qc = t;
    }

    // y[b][s][l] = sum_h xc[s][b][h] * P[s][h][l] * rsqrt(tr)
    // C row stride = SEQ*HID (b-major output), batch stride = HID (s offset)
    gemm256_wmma<<<gg, 256, 0, stream>>>(xc, 0, Pc, out,
                                         (long long)SEQ * HID, (long long)HID,
                                         nullptr, 1.0f, 0.f, rsqTr);

    minred_kernel<<<1, 256, 0, stream>>>(minE, out + (size_t)BATCH * SEQ * HID);
}